// TemporalNMF_27960237097126
// MI455X (gfx1250) — compile-verified
//
#include <hip/hip_runtime.h>
#include <hip/hip_bf16.h>
#include <math.h>

typedef __attribute__((ext_vector_type(2))) float v2f;
typedef __attribute__((ext_vector_type(4))) float v4f;
typedef __attribute__((ext_vector_type(8))) float v8f;

#define LEAK 0.3f
constexpr int D = 64, B = 256, NN = 10000, E = 32, F = 12, H = 128, M = 4;
constexpr int S = D * B;                 // 16384 samples
constexpr int LROW = H + 4;              // padded LDS row stride (bank-conflict-free)

// ---- d_out layout (floats) ----
constexpr long long OUT_REC_AGE = 0;
constexpr long long OUT_REC_EMB = (long long)D * B * B * M;          // 16777216
constexpr long long OUT_FAC_AGE = OUT_REC_EMB * 2;                    // 33554432
constexpr long long OUT_FAC_EMB = OUT_FAC_AGE + (long long)D * B * F;
constexpr long long OUT_OFFSETS = OUT_FAC_EMB + (long long)D * B * F;
constexpr long long OUT_EMBS    = OUT_OFFSETS + (long long)D * B * F;

// ---- workspace layout (floats) ----
constexpr long long WS_STATS = 0;                     // mean, inv_std
constexpr long long WS_W1A   = 16;
constexpr long long WS_W1O   = WS_W1A + H;
constexpr long long WS_W2A   = WS_W1O + H;
constexpr long long WS_W3A   = WS_W2A + (long long)H * H;
constexpr long long WS_W4A   = WS_W3A + (long long)H * H;
constexpr long long WS_W2O   = WS_W4A + (long long)H * H;
constexpr long long WS_W3O   = WS_W2O + (long long)H * H;
constexpr long long WS_W4O   = WS_W3O + (long long)H * H;
constexpr long long WS_A     = WS_W4O + (long long)H * H;
constexpr long long WS_X     = WS_A + S;
constexpr long long WS_Y     = WS_X + (long long)S * H;

__device__ inline v8f wmma4(v2f a, v2f b, v8f c) {
  // D = A(16x4 f32) * B(4x16 f32) + C(16x16 f32)
  return __builtin_amdgcn_wmma_f32_16x16x4_f32(false, a, false, b, (short)0, c,
                                               false, false);
}

__device__ inline v8f vzero8() {
  v8f z;
#pragma unroll
  for (int i = 0; i < 8; ++i) z[i] = 0.f;
  return z;
}

// ---------------- stats over positive ages ----------------
__global__ __launch_bounds__(256) void stats_kernel(const float* __restrict__ ages,
                                                    float* __restrict__ stats) {
  __shared__ float s_sum[256], s_sq[256];
  __shared__ int s_cnt[256];
  int t = threadIdx.x;
  float sum = 0.f, sq = 0.f;
  int cnt = 0;
  for (int i = t; i < D * NN; i += 256) {
    float x = ages[i];
    if (x > 0.f) { sum += x; sq += x * x; ++cnt; }
  }
  s_sum[t] = sum; s_sq[t] = sq; s_cnt[t] = cnt;
  __syncthreads();
  for (int o = 128; o > 0; o >>= 1) {
    if (t < o) { s_sum[t] += s_sum[t + o]; s_sq[t] += s_sq[t + o]; s_cnt[t] += s_cnt[t + o]; }
    __syncthreads();
  }
  if (t == 0) {
    float c = (float)s_cnt[0];
    float mean = s_sum[0] / c;
    float var = (s_sq[0] - c * mean * mean) / (c - 1.f);  // unbiased (ddof=1)
    stats[0] = mean;
    stats[1] = rsqrtf(var);
  }
}

// ---------------- weight-norm: w = v * g / ||v||_2 per row ----------------
__global__ __launch_bounds__(128) void wn_kernel(const float* __restrict__ v,
                                                 const float* __restrict__ g,
                                                 float* __restrict__ w) {
  __shared__ float red[128];
  int row = blockIdx.x, t = threadIdx.x;
  float x = v[row * H + t];
  red[t] = x * x;
  __syncthreads();
  for (int o = 64; o > 0; o >>= 1) {
    if (t < o) red[t] += red[t + o];
    __syncthreads();
  }
  float scale = g[row] * rsqrtf(red[0]);
  w[row * H + t] = x * scale;
}

// layer-1 (in_dim==1): ||v_row|| == |v|, so w = v*g/|v|
__global__ __launch_bounds__(256) void l1prep_kernel(
    const float* __restrict__ v1a, const float* __restrict__ g1a,
    const float* __restrict__ v1o, const float* __restrict__ g1o,
    float* __restrict__ w1a, float* __restrict__ w1o) {
  int t = threadIdx.x;
  if (t < H) {
    w1a[t] = v1a[t] * g1a[t] / fabsf(v1a[t]);
  } else {
    int j = t - H;
    w1o[j] = v1o[j] * g1o[j] / fabsf(v1o[j]);
  }
}

// ---------------- embedding gather + L1 max_norm renorm + abs ----------------
__global__ __launch_bounds__(32) void embs_kernel(const float* __restrict__ emb,
                                                  const int* __restrict__ idxs,
                                                  float* __restrict__ out) {
  int b = blockIdx.x, t = threadIdx.x;
  float v = emb[(long long)idxs[b] * E + t];
  float s = fabsf(v);
  for (int o = 16; o > 0; o >>= 1) s += __shfl_xor(s, o);
  float scale = (s > 1.0f) ? (1.0f / (s + 1e-7f)) : 1.0f;
  out[b * E + t] = fabsf(v * scale);
}

// ---------------- a[d,b] = (ages[d, idxs[b]] - mean) * inv_std ----------------
__global__ __launch_bounds__(256) void a_kernel(const float* __restrict__ ages,
                                                const int* __restrict__ idxs,
                                                const float* __restrict__ stats,
                                                float* __restrict__ a) {
  int d = blockIdx.x, b = threadIdx.x;
  float x = ages[d * NN + idxs[b]];
  a[d * B + b] = (x - stats[0]) * stats[1];
}

// ---------------- layer 1: h = lrelu(a * w1 + b1), out [S,H] ----------------
__global__ __launch_bounds__(256) void l1_kernel(const float* __restrict__ a,
                                                 const float* __restrict__ w1,
                                                 const float* __restrict__ b1,
                                                 float* __restrict__ out) {
  int g = blockIdx.x * 256 + threadIdx.x;
  int s = g >> 7, j = g & (H - 1);
  float h = fmaf(a[s], w1[j], b1[j]);
  out[g] = h > 0.f ? h : LEAK * h;
}

// ---------------- hidden layer GEMM: Y = lrelu(X @ W^T + b), [S,H]x[H,H] ----------------
__global__ __launch_bounds__(256) void gemm_hidden(const float* __restrict__ X,
                                                   const float* __restrict__ W,
                                                   const float* __restrict__ bias,
                                                   float* __restrict__ Y) {
  extern __shared__ float sm[];
  float* Xs = sm;                    // [128][LROW] row-major
  float* Ws = sm + 128 * LROW;       // [128][LROW] row-major (W[n][k])
  const int t = threadIdx.x;
  const int m0 = blockIdx.x * 128;

  for (int idx = t; idx < 128 * H; idx += 256) {
    int r = idx >> 7, c = idx & (H - 1);
    Xs[r * LROW + c] = X[(long long)(m0 + r) * H + c];
    Ws[r * LROW + c] = W[idx];
  }
  __syncthreads();

  const int wave = t >> 5, lane = t & 31;
  const int half = lane >> 4, l16 = lane & 15;
  const int mloc = wave * 16 + l16;

  v8f acc[8];
#pragma unroll
  for (int i = 0; i < 8; ++i) acc[i] = vzero8();

  for (int k0 = 0; k0 < H; k0 += 4) {
    v2f a;
    a.x = Xs[mloc * LROW + k0 + 2 * half];
    a.y = Xs[mloc * LROW + k0 + 2 * half + 1];
#pragma unroll
    for (int nt = 0; nt < 8; ++nt) {
      int n = nt * 16 + l16;
      v2f b;
      b.x = Ws[n * LROW + k0 + 2 * half];
      b.y = Ws[n * LROW + k0 + 2 * half + 1];
      acc[nt] = wmma4(a, b, acc[nt]);
    }
  }

#pragma unroll
  for (int nt = 0; nt < 8; ++nt) {
    int n = nt * 16 + l16;
    float bn = bias[n];
#pragma unroll
    for (int r = 0; r < 8; ++r) {
      int row = m0 + wave * 16 + r + 8 * half;
      float v = acc[nt][r] + bn;
      Y[(long long)row * H + n] = v > 0.f ? v : LEAK * v;
    }
  }
}

// ---------------- final age layer: out[S,F] = X @ Wl^T + bl ----------------
__global__ __launch_bounds__(256) void final_age_kernel(const float* __restrict__ X,
                                                        const float* __restrict__ Wl,
                                                        const float* __restrict__ bl,
                                                        float* __restrict__ out) {
  extern __shared__ float sm[];
  float* Xs = sm;                 // [128][LROW]
  float* Ws = sm + 128 * LROW;    // [16][LROW], rows >= F zero-padded
  const int t = threadIdx.x;
  const int m0 = blockIdx.x * 128;

  for (int idx = t; idx < 128 * H; idx += 256) {
    int r = idx >> 7, c = idx & (H - 1);
    Xs[r * LROW + c] = X[(long long)(m0 + r) * H + c];
  }
  for (int idx = t; idx < 16 * H; idx += 256) {
    int r = idx >> 7, c = idx & (H - 1);
    Ws[r * LROW + c] = (r < F) ? Wl[r * H + c] : 0.f;
  }
  __syncthreads();

  const int wave = t >> 5, lane = t & 31;
  const int half = lane >> 4, l16 = lane & 15;
  const int mloc = wave * 16 + l16;

  v8f acc = vzero8();
  for (int k0 = 0; k0 < H; k0 += 4) {
    v2f a, b;
    a.x = Xs[mloc * LROW + k0 + 2 * half];
    a.y = Xs[mloc * LROW + k0 + 2 * half + 1];
    b.x = Ws[l16 * LROW + k0 + 2 * half];
    b.y = Ws[l16 * LROW + k0 + 2 * half + 1];
    acc = wmma4(a, b, acc);
  }
  if (l16 < F) {
    float bn = bl[l16];
#pragma unroll
    for (int r = 0; r < 8; ++r) {
      int row = m0 + wave * 16 + r + 8 * half;
      out[(long long)row * F + l16] = acc[r] + bn;
    }
  }
}

// ------- final offset layer fused with E-contraction:
// basis[s, e*F+f] = X@Wl^T + bl ; offsets[s,f] += basis * embs[s%B, e]
__global__ __launch_bounds__(256) void final_off_kernel(
    const float* __restrict__ X, const float* __restrict__ Wl,
    const float* __restrict__ bl, const float* __restrict__ embs,
    const float* __restrict__ fac_age, float* __restrict__ offs_out,
    float* __restrict__ fac_emb_out) {
  extern __shared__ float sm[];
  float* Xs   = sm;                              // [128][LROW]
  float* Ws   = sm + 128 * LROW;                 // [16][LROW] per n-tile
  float* offs = sm + 128 * LROW + 16 * LROW;     // [128][F]
  const int t = threadIdx.x;
  const int m0 = blockIdx.x * 128;

  for (int idx = t; idx < 128 * H; idx += 256) {
    int r = idx >> 7, c = idx & (H - 1);
    Xs[r * LROW + c] = X[(long long)(m0 + r) * H + c];
  }
  for (int idx = t; idx < 128 * F; idx += 256) offs[idx] = 0.f;
  __syncthreads();

  const int wave = t >> 5, lane = t & 31;
  const int half = lane >> 4, l16 = lane & 15;
  const int mloc = wave * 16 + l16;

  for (int nt = 0; nt < (E * F) / 16; ++nt) {   // 24 tiles of 16 output cols
    for (int idx = t; idx < 16 * H; idx += 256) {
      int r = idx >> 7, c = idx & (H - 1);
      Ws[r * LROW + c] = Wl[(long long)(nt * 16 + r) * H + c];
    }
    __syncthreads();

    v8f acc = vzero8();
    for (int k0 = 0; k0 < H; k0 += 4) {
      v2f a, b;
      a.x = Xs[mloc * LROW + k0 + 2 * half];
      a.y = Xs[mloc * LROW + k0 + 2 * half + 1];
      b.x = Ws[l16 * LROW + k0 + 2 * half];
      b.y = Ws[l16 * LROW + k0 + 2 * half + 1];
      acc = wmma4(a, b, acc);
    }

    int n = nt * 16 + l16;
    int e = n / F, f = n - e * F;
    float bn = bl[n];
#pragma unroll
    for (int r = 0; r < 8; ++r) {
      int rloc = wave * 16 + r + 8 * half;
      int s = m0 + rloc;
      float ev = embs[(s & (B - 1)) * E + e];
      atomicAdd(&offs[rloc * F + f], (acc[r] + bn) * ev);
    }
    __syncthreads();
  }

  for (int idx = t; idx < 128 * F; idx += 256) {
    int rloc = idx / F, f = idx - rloc * F;
    long long s = m0 + rloc;
    float ov = offs[idx];
    offs_out[s * F + f] = ov;
    fac_emb_out[s * F + f] = fac_age[s * F + f] + ov;
  }
}

// ------- reconstruction: rec[d,i,j,m] = sum_f relu(fac)[d,i,f]*relu(fac)[d,j,f]*om[f,m]
__global__ __launch_bounds__(256) void recon_kernel(const float* __restrict__ fac,
                                                    const float* __restrict__ om_raw,
                                                    float* __restrict__ rec) {
  __shared__ float Gs[B * 17];        // [256][17], relu(factors)
  __shared__ float om_s[F * M];       // softplus-normalized map
  __shared__ float cn[M];
  const int d = blockIdx.x >> 1;
  const int iBase = (blockIdx.x & 1) * 128;
  const int t = threadIdx.x;

  if (t < F * M) {
    float x = om_raw[t];
    om_s[t] = (x > 20.f) ? x : log1pf(expf(x));   // softplus
  }
  __syncthreads();
  if (t < M) {
    float cs = 0.f;
    for (int f = 0; f < F; ++f) cs += om_s[f * M + t];
    cn[t] = (float)F / cs;                        // column-normalize * num_factors
  }
  __syncthreads();
  if (t < F * M) om_s[t] *= cn[t & (M - 1)];
  for (int idx = t; idx < B * F; idx += 256) {
    int j = idx / F, f = idx - j * F;
    float x = fac[((long long)d * B + j) * F + f];
    Gs[j * 17 + f] = x > 0.f ? x : 0.f;
  }
  __syncthreads();

  const int wave = t >> 5, lane = t & 31;
  const int half = lane >> 4, l16 = lane & 15;
  const int iloc = iBase + wave * 16 + l16;

  for (int jt = 0; jt < 16; ++jt) {
    v8f acc[M];
#pragma unroll
    for (int m = 0; m < M; ++m) acc[m] = vzero8();

#pragma unroll
    for (int k0 = 0; k0 < F; k0 += 4) {           // K = 12 -> 3 WMMA steps
      int ka = k0 + 2 * half;
      v2f a;
      a.x = Gs[iloc * 17 + ka];
      a.y = Gs[iloc * 17 + ka + 1];
      float g0 = Gs[(jt * 16 + l16) * 17 + ka];
      float g1 = Gs[(jt * 16 + l16) * 17 + ka + 1];
#pragma unroll
      for (int m = 0; m < M; ++m) {
        v2f b;
        b.x = g0 * om_s[ka * M + m];
        b.y = g1 * om_s[(ka + 1) * M + m];
        acc[m] = wmma4(a, b, acc[m]);
      }
    }

#pragma unroll
    for (int r = 0; r < 8; ++r) {
      int i = iBase + wave * 16 + r + 8 * half;
      int j = jt * 16 + l16;
      v4f o4;
      o4.x = acc[0][r]; o4.y = acc[1][r]; o4.z = acc[2][r]; o4.w = acc[3][r];
      *(v4f*)&rec[(((long long)d * B + i) * B + j) * M] = o4;
    }
  }
}

extern "C" void kernel_launch(void* const* d_in, const int* in_sizes, int n_in,
                              void* d_out, int out_size, void* d_ws, size_t ws_size,
                              hipStream_t stream) {
  (void)in_sizes; (void)n_in; (void)out_size; (void)ws_size;
  const float* ages = (const float*)d_in[0];
  const int*   idxs = (const int*)d_in[1];
  const float* emb  = (const float*)d_in[2];
  const float* om   = (const float*)d_in[3];
  // age_mlp: 4 weight-normed layers (v,g,b) + (w,b)
  const float* v1a = (const float*)d_in[4];  const float* g1a = (const float*)d_in[5];  const float* b1a = (const float*)d_in[6];
  const float* v2a = (const float*)d_in[7];  const float* g2a = (const float*)d_in[8];  const float* b2a = (const float*)d_in[9];
  const float* v3a = (const float*)d_in[10]; const float* g3a = (const float*)d_in[11]; const float* b3a = (const float*)d_in[12];
  const float* v4a = (const float*)d_in[13]; const float* g4a = (const float*)d_in[14]; const float* b4a = (const float*)d_in[15];
  const float* wLa = (const float*)d_in[16]; const float* bLa = (const float*)d_in[17];
  // off_mlp
  const float* v1o = (const float*)d_in[18]; const float* g1o = (const float*)d_in[19]; const float* b1o = (const float*)d_in[20];
  const float* v2o = (const float*)d_in[21]; const float* g2o = (const float*)d_in[22]; const float* b2o = (const float*)d_in[23];
  const float* v3o = (const float*)d_in[24]; const float* g3o = (const float*)d_in[25]; const float* b3o = (const float*)d_in[26];
  const float* v4o = (const float*)d_in[27]; const float* g4o = (const float*)d_in[28]; const float* b4o = (const float*)d_in[29];
  const float* wLo = (const float*)d_in[30]; const float* bLo = (const float*)d_in[31];

  float* out = (float*)d_out;
  float* ws  = (float*)d_ws;

  stats_kernel<<<1, 256, 0, stream>>>(ages, ws + WS_STATS);
  wn_kernel<<<H, H, 0, stream>>>(v2a, g2a, ws + WS_W2A);
  wn_kernel<<<H, H, 0, stream>>>(v3a, g3a, ws + WS_W3A);
  wn_kernel<<<H, H, 0, stream>>>(v4a, g4a, ws + WS_W4A);
  wn_kernel<<<H, H, 0, stream>>>(v2o, g2o, ws + WS_W2O);
  wn_kernel<<<H, H, 0, stream>>>(v3o, g3o, ws + WS_W3O);
  wn_kernel<<<H, H, 0, stream>>>(v4o, g4o, ws + WS_W4O);
  l1prep_kernel<<<1, 256, 0, stream>>>(v1a, g1a, v1o, g1o, ws + WS_W1A, ws + WS_W1O);
  embs_kernel<<<B, 32, 0, stream>>>(emb, idxs, out + OUT_EMBS);
  a_kernel<<<D, B, 0, stream>>>(ages, idxs, ws + WS_STATS, ws + WS_A);

  float* Xb = ws + WS_X;
  float* Yb = ws + WS_Y;
  const size_t sm_gemm  = (size_t)(2 * 128 * LROW) * sizeof(float);
  const size_t sm_final = (size_t)(128 * LROW + 16 * LROW) * sizeof(float);
  const size_t sm_off   = (size_t)(128 * LROW + 16 * LROW + 128 * F) * sizeof(float);

  // ---- age MLP chain ----
  l1_kernel<<<(S * H) / 256, 256, 0, stream>>>(ws + WS_A, ws + WS_W1A, b1a, Xb);
  gemm_hidden<<<S / 128, 256, sm_gemm, stream>>>(Xb, ws + WS_W2A, b2a, Yb);
  gemm_hidden<<<S / 128, 256, sm_gemm, stream>>>(Yb, ws + WS_W3A, b3a, Xb);
  gemm_hidden<<<S / 128, 256, sm_gemm, stream>>>(Xb, ws + WS_W4A, b4a, Yb);
  final_age_kernel<<<S / 128, 256, sm_final, stream>>>(Yb, wLa, bLa, out + OUT_FAC_AGE);

  // ---- offsetter MLP chain + fused E-contraction ----
  l1_kernel<<<(S * H) / 256, 256, 0, stream>>>(ws + WS_A, ws + WS_W1O, b1o, Xb);
  gemm_hidden<<<S / 128, 256, sm_gemm, stream>>>(Xb, ws + WS_W2O, b2o, Yb);
  gemm_hidden<<<S / 128, 256, sm_gemm, stream>>>(Yb, ws + WS_W3O, b3o, Xb);
  gemm_hidden<<<S / 128, 256, sm_gemm, stream>>>(Xb, ws + WS_W4O, b4o, Yb);
  final_off_kernel<<<S / 128, 256, sm_off, stream>>>(
      Yb, wLo, bLo, out + OUT_EMBS, out + OUT_FAC_AGE,
      out + OUT_OFFSETS, out + OUT_FAC_EMB);

  // ---- reconstructions ----
  recon_kernel<<<D * 2, 256, 0, stream>>>(out + OUT_FAC_AGE, om, out + OUT_REC_AGE);
  recon_kernel<<<D * 2, 256, 0, stream>>>(out + OUT_FAC_EMB, om, out + OUT_REC_EMB);
}